// IDEF_42511586296385
// MI455X (gfx1250) — compile-verified
//
#include <hip/hip_runtime.h>
#include <math.h>

// ---------------------------------------------------------------------------
// Problem constants (from reference)
// ---------------------------------------------------------------------------
#define NDIM    256
#define HID     1024
#define TGRID   512
#define SREAL   1000      // MC_SAMPLES
#define SPAD    1024      // padded sample count for 16x16 WMMA tiling

typedef float v2f __attribute__((ext_vector_type(2)));
typedef float v8f __attribute__((ext_vector_type(8)));

// ---------------------------------------------------------------------------
// Workspace layout (floats)
// ---------------------------------------------------------------------------
#define WS_M      0                               // 512*256 spline M
#define WS_ZINT   (WS_M + TGRID*NDIM)             // 1024*256
#define WS_CS     (WS_ZINT + SPAD*NDIM)           // 1024
#define WS_T      (WS_CS + SPAD)                  // 1024*1024 (T, later E in place)
#define WS_F      (WS_T + SPAD*HID)               // 1024*256
#define WS_G      (WS_F + SPAD*NDIM)              // 256*1024
#define WS_Q      (WS_G + NDIM*HID)               // 256
#define WS_V      (WS_Q + NDIM)                   // 256
#define WS_W      (WS_V + NDIM)                   // 1024
#define WS_P      (WS_W + HID)                    // 1024
#define WS_R      (WS_P + HID)                    // 1024
#define WS_C      (WS_R + HID)                    // 1
#define WS_UI     (WS_C + 1)                      // 1024

// ---------------------------------------------------------------------------
// 1) Natural cubic spline second derivatives M (Thomas algorithm).
// ---------------------------------------------------------------------------
__global__ void spline_coeffs_kernel(const float* __restrict__ z,
                                     const float* __restrict__ t,
                                     float* __restrict__ Mbuf) {
    __shared__ float hs[TGRID];      // h[i] = t[i+1]-t[i], i<511
    __shared__ float cp[TGRID];      // Thomas c' coefficients
    __shared__ float invden[TGRID];  // 1/denominator per interior row
    const int d = threadIdx.x;       // 0..255

    for (int i = d; i < TGRID - 1; i += NDIM) hs[i] = t[i + 1] - t[i];
    __syncthreads();

    if (d == 0) {
        float cprev = 0.0f;
        for (int i = 1; i <= TGRID - 2; ++i) {
            float lo = hs[i - 1], up = hs[i];
            float diag = 2.0f * (hs[i - 1] + hs[i]);
            float den = diag - lo * cprev;
            float inv = 1.0f / den;
            cprev = up * inv;
            cp[i] = cprev;
            invden[i] = inv;
        }
    }
    __syncthreads();

    // forward sweep: dp stored into Mbuf rows
    float dprev = 0.0f;
    for (int i = 1; i <= TGRID - 2; ++i) {
        float sl1 = (z[(i + 1) * NDIM + d] - z[i * NDIM + d]) / hs[i];
        float sl0 = (z[i * NDIM + d] - z[(i - 1) * NDIM + d]) / hs[i - 1];
        float rhs = 6.0f * (sl1 - sl0);
        dprev = (rhs - hs[i - 1] * dprev) * invden[i];
        Mbuf[i * NDIM + d] = dprev;
    }
    // backward substitution (natural BC: M[0]=M[T-1]=0)
    float Mnext = 0.0f;
    Mbuf[(TGRID - 1) * NDIM + d] = 0.0f;
    for (int i = TGRID - 2; i >= 1; --i) {
        float Mi = Mbuf[i * NDIM + d] - cp[i] * Mnext;
        Mbuf[i * NDIM + d] = Mi;
        Mnext = Mi;
    }
    Mbuf[0 * NDIM + d] = 0.0f;
}

// ---------------------------------------------------------------------------
// 2) Spline evaluation at MC abscissae s = t_i * u; c_s = exp(-(t_i-s)^2).
//    Padded rows (s>=1000) zeroed so they vanish from all reductions.
// ---------------------------------------------------------------------------
__global__ void spline_eval_kernel(const float* __restrict__ z,
                                   const float* __restrict__ t,
                                   const float* __restrict__ Mbuf,
                                   const float* __restrict__ u,
                                   const float* __restrict__ t_i,
                                   float* __restrict__ Zint,
                                   float* __restrict__ csamp) {
    const int s = blockIdx.x;        // 0..1023
    const int d = threadIdx.x;       // 0..255
    if (s >= SREAL) {
        Zint[(size_t)s * NDIM + d] = 0.0f;
        if (d == 0) csamp[s] = 0.0f;
        return;
    }
    const float ub = t_i[0];
    const float sv = ub * u[s];
    // searchsorted(t, sv, side='right') - 1, clipped to [0, T-2]
    int lo = 0, hi = TGRID;
    while (lo < hi) { int mid = (lo + hi) >> 1; if (t[mid] <= sv) lo = mid + 1; else hi = mid; }
    int idx = lo - 1;
    if (idx < 0) idx = 0;
    if (idx > TGRID - 2) idx = TGRID - 2;

    const float hstep = t[idx + 1] - t[idx];
    const float dt = sv - t[idx];
    const float y0 = z[idx * NDIM + d], y1 = z[(idx + 1) * NDIM + d];
    const float m0 = Mbuf[idx * NDIM + d], m1 = Mbuf[(idx + 1) * NDIM + d];
    float val = y0
              + dt * ((y1 - y0) / hstep - hstep * (2.0f * m0 + m1) * (1.0f / 6.0f))
              + dt * dt * m0 * 0.5f
              + dt * dt * dt * (m1 - m0) / (6.0f * hstep);
    Zint[(size_t)s * NDIM + d] = val;
    if (d == 0) { float df = ub - sv; csamp[s] = expf(-df * df); }
}

// ---------------------------------------------------------------------------
// 3) WMMA f32 GEMM: D = act(A·B + bias), strip-mined.
//    Each wave owns a 64x16 output strip (4 M-tiles x 1 N-tile): per K-step one
//    B fragment feeds 4 V_WMMA_F32_16X16X4_F32, quadrupling WMMA density and
//    B reuse. All dims/strides are template constants so global loads fold to
//    base + 24-bit immediate offsets (no per-load 64-bit address chains).
//    Fragment layouts per CDNA5 ISA §7.12.2:
//      A 16x4:  lane L -> m=L%16, k in {k0+2*(L/16), k0+2*(L/16)+1}
//      B 4x16:  lane L -> n=L%16, same k pair
//      C/D:     vgpr r -> row 16*tile + r + 8*(L/16), col 16*tj + L%16
// ---------------------------------------------------------------------------
template <bool TRANS_A, bool HAS_BIAS, bool TANH_ACT,
          int M, int N, int K, int LDA, int LDB, int LDD>
__global__ void wmma_gemm_f32_kernel(const float* __restrict__ A,
                                     const float* __restrict__ B,
                                     const float* __restrict__ bias,
                                     float* __restrict__ D) {
    const int lane = threadIdx.x & 31;
    const int wave = threadIdx.x >> 5;
    constexpr int STRIPS_N = N >> 4;          // 16-wide N tiles
    constexpr int STRIPS_M = M >> 6;          // 64-tall M strips
    const int strip = blockIdx.x * (blockDim.x >> 5) + wave;
    if (strip >= STRIPS_M * STRIPS_N) return; // uniform per wave -> EXEC all-ones
    const int si = strip / STRIPS_N;          // M strip index (64 rows)
    const int sj = strip % STRIPS_N;          // N tile index (16 cols)

    const int mrow = lane & 15;
    const int kp = (lane >> 4) << 1;          // 0 or 2
    const int col = sj * 16 + (lane & 15);
    const int row0 = si * 64 + mrow;

    v8f acc0 = {}, acc1 = {}, acc2 = {}, acc3 = {};

#pragma unroll 2
    for (int k0 = 0; k0 < K; k0 += 4) {
        const int ka = k0 + kp;
        v2f b;
        b.x = B[(size_t)ka * LDB + col];
        b.y = B[(size_t)(ka + 1) * LDB + col];

        v2f a0, a1, a2, a3;
        if (TRANS_A) {
            const float* Ar0 = A + (size_t)ka * LDA + row0;
            const float* Ar1 = A + (size_t)(ka + 1) * LDA + row0;
            a0.x = Ar0[0];  a0.y = Ar1[0];
            a1.x = Ar0[16]; a1.y = Ar1[16];
            a2.x = Ar0[32]; a2.y = Ar1[32];
            a3.x = Ar0[48]; a3.y = Ar1[48];
        } else {
            const float* Ar = A + (size_t)row0 * LDA + ka;
            a0.x = Ar[0 * 16 * LDA];     a0.y = Ar[0 * 16 * LDA + 1];
            a1.x = Ar[1 * 16 * LDA];     a1.y = Ar[1 * 16 * LDA + 1];
            a2.x = Ar[2 * 16 * LDA];     a2.y = Ar[2 * 16 * LDA + 1];
            a3.x = Ar[3 * 16 * LDA];     a3.y = Ar[3 * 16 * LDA + 1];
        }
        acc0 = __builtin_amdgcn_wmma_f32_16x16x4_f32(false, a0, false, b, (short)0, acc0, false, false);
        acc1 = __builtin_amdgcn_wmma_f32_16x16x4_f32(false, a1, false, b, (short)0, acc1, false, false);
        acc2 = __builtin_amdgcn_wmma_f32_16x16x4_f32(false, a2, false, b, (short)0, acc2, false, false);
        acc3 = __builtin_amdgcn_wmma_f32_16x16x4_f32(false, a3, false, b, (short)0, acc3, false, false);
    }

    const float bv = HAS_BIAS ? bias[col] : 0.0f;
    const int rbase = si * 64 + ((lane >> 4) << 3);
#pragma unroll
    for (int m = 0; m < 4; ++m) {
        const v8f acc = (m == 0) ? acc0 : (m == 1) ? acc1 : (m == 2) ? acc2 : acc3;
#pragma unroll
        for (int r = 0; r < 8; ++r) {
            const int row = rbase + m * 16 + r;
            float val = acc[r] + bv;
            if (TANH_ACT) val = tanhf(val);
            D[(size_t)row * LDD + col] = val;
        }
    }
}

// ---------------------------------------------------------------------------
// 4) Column reductions over T: p[h]=sum_s c_s*T, r[h]=sum_s E; T <- E in place.
// ---------------------------------------------------------------------------
__global__ void pr_transform_kernel(float* __restrict__ T,
                                    const float* __restrict__ csamp,
                                    float* __restrict__ p,
                                    float* __restrict__ r) {
    const int h = blockIdx.x * blockDim.x + threadIdx.x;   // 0..1023
    float pt = 0.0f, rt = 0.0f;
    for (int s = 0; s < SPAD; ++s) {
        const float c = csamp[s];
        const float tv = T[(size_t)s * HID + h];
        pt += c * tv;
        const float e = c * (1.0f - tv * tv);
        rt += e;
        T[(size_t)s * HID + h] = e;
    }
    p[h] = pt;
    r[h] = rt;
}

// 5) q[d] = sum_s c_s*F[s,d];  C = sum_s c_s
__global__ void q_reduce_kernel(const float* __restrict__ F,
                                const float* __restrict__ csamp,
                                float* __restrict__ q,
                                float* __restrict__ Cred) {
    const int d = threadIdx.x;
    float qt = 0.0f;
    for (int s = 0; s < SPAD; ++s) qt += csamp[s] * F[(size_t)s * NDIM + d];
    q[d] = qt;
    if (d == 0) {
        float c = 0.0f;
        for (int s = 0; s < SPAD; ++s) c += csamp[s];
        *Cred = c;
    }
}

// 6) v[d] = sum_n g[n]*A[n,d]
__global__ void v_kernel(const float* __restrict__ Amat,
                         const float* __restrict__ g,
                         float* __restrict__ v) {
    const int d = threadIdx.x;
    float acc = 0.0f;
    for (int n = 0; n < NDIM; ++n) acc += g[n] * Amat[(size_t)n * NDIM + d];
    v[d] = acc;
}

// 7) w[h] = sum_d W2[h,d]*v[d]
__global__ void w_kernel(const float* __restrict__ W2,
                         const float* __restrict__ v,
                         float* __restrict__ w) {
    const int h = blockIdx.x * blockDim.x + threadIdx.x;
    float acc = 0.0f;
    for (int d = 0; d < NDIM; ++d) acc += W2[(size_t)h * NDIM + d] * v[d];
    w[h] = acc;
}

// 8) u_i[h] = (1 - tanh^2(z_i·W1 + b1)[h]) * w[h]    (for adfdz)
__global__ void ui_kernel(const float* __restrict__ z_i,
                          const float* __restrict__ W1,
                          const float* __restrict__ b1,
                          const float* __restrict__ w,
                          float* __restrict__ ui) {
    const int h = blockIdx.x * blockDim.x + threadIdx.x;
    float acc = b1[h];
    for (int d = 0; d < NDIM; ++d) acc += z_i[d] * W1[(size_t)d * HID + h];
    const float th = tanhf(acc);
    ui[h] = (1.0f - th * th) * w[h];
}

// ---------------------------------------------------------------------------
// 9) Finalize: assemble all outputs.
//    [0,256)  f_out = (L/1000) * A·q
//    [256,512) adfdz[k] = sum_h W1[k,h]*ui[h]
//    [512, 512+591104) param grads scaled by L
//    last element: a_int_dfdt = 0
// ---------------------------------------------------------------------------
__global__ void finalize_kernel(const float* __restrict__ Amat,
                                const float* __restrict__ W1,
                                const float* __restrict__ g,
                                const float* __restrict__ t_i,
                                const float* __restrict__ q,
                                const float* __restrict__ v,
                                const float* __restrict__ w,
                                const float* __restrict__ p,
                                const float* __restrict__ r,
                                const float* __restrict__ Cred,
                                const float* __restrict__ G,
                                const float* __restrict__ ui,
                                float* __restrict__ out, int out_size) {
    const int i = blockIdx.x * blockDim.x + threadIdx.x;
    if (i >= out_size) return;
    const float L = t_i[0];

    if (i < NDIM) {                                   // f_out
        float acc = 0.0f;
        const float* Ar = Amat + (size_t)i * NDIM;
        for (int d = 0; d < NDIM; ++d) acc += Ar[d] * q[d];
        out[i] = (L * acc) * (1.0f / (float)SREAL);
    } else if (i < 2 * NDIM) {                        // adfdz
        const int k = i - NDIM;
        float acc = 0.0f;
        for (int h = 0; h < HID; ++h) acc += W1[(size_t)k * HID + h] * ui[h];
        out[i] = acc;
    } else if (i == out_size - 1) {                   // a_int_dfdt
        out[i] = 0.0f;
    } else {                                          // a_int_dfdp
        const int j = i - 2 * NDIM;
        float val;
        if (j < NDIM * HID) {                         // W1 grad (256,1024)
            const int d = j >> 10, h = j & (HID - 1);
            val = G[(size_t)d * HID + h] * w[h];
        } else if (j < NDIM * HID + HID) {            // b1 grad
            const int h = j - NDIM * HID;
            val = r[h] * w[h];
        } else if (j < 2 * NDIM * HID + HID) {        // W2 grad (1024,256)
            const int jj = j - (NDIM * HID + HID);
            const int h = jj >> 8, d = jj & (NDIM - 1);
            val = p[h] * v[d];
        } else if (j < 2 * NDIM * HID + HID + NDIM) { // b2 grad
            const int d = j - (2 * NDIM * HID + HID);
            val = (*Cred) * v[d];
        } else {                                      // A grad (256,256)
            const int jj = j - (2 * NDIM * HID + HID + NDIM);
            const int n = jj >> 8, d = jj & (NDIM - 1);
            val = g[n] * q[d];
        }
        out[i] = L * val;
    }
}

// ---------------------------------------------------------------------------
// Launcher
// ---------------------------------------------------------------------------
extern "C" void kernel_launch(void* const* d_in, const int* in_sizes, int n_in,
                              void* d_out, int out_size, void* d_ws, size_t ws_size,
                              hipStream_t stream) {
    const float* z_i = (const float*)d_in[0];
    const float* t_i = (const float*)d_in[1];
    const float* z   = (const float*)d_in[2];
    const float* t   = (const float*)d_in[3];
    const float* g   = (const float*)d_in[4];
    const float* u   = (const float*)d_in[5];
    const float* W1  = (const float*)d_in[6];
    const float* b1  = (const float*)d_in[7];
    const float* W2  = (const float*)d_in[8];
    const float* b2  = (const float*)d_in[9];
    const float* A   = (const float*)d_in[10];
    float* out = (float*)d_out;

    float* ws   = (float*)d_ws;
    float* Mbuf = ws + WS_M;
    float* Zint = ws + WS_ZINT;
    float* cs   = ws + WS_CS;
    float* T    = ws + WS_T;     // becomes E in place after pr_transform
    float* F    = ws + WS_F;
    float* G    = ws + WS_G;
    float* q    = ws + WS_Q;
    float* v    = ws + WS_V;
    float* w    = ws + WS_W;
    float* p    = ws + WS_P;
    float* r    = ws + WS_R;
    float* Cr   = ws + WS_C;
    float* ui   = ws + WS_UI;

    // 1) spline coefficients
    spline_coeffs_kernel<<<1, NDIM, 0, stream>>>(z, t, Mbuf);
    // 2) spline eval + MC kernel weights
    spline_eval_kernel<<<SPAD, NDIM, 0, stream>>>(z, t, Mbuf, u, t_i, Zint, cs);

    // 3) GEMM1: T = tanh(Zint(1024x256) · W1(256x1024) + b1)
    //    strips = (1024/64)*(1024/16) = 1024 -> 256 blocks x 4 waves
    wmma_gemm_f32_kernel<false, true, true, SPAD, HID, NDIM, NDIM, HID, HID>
        <<<256, 128, 0, stream>>>(Zint, W1, b1, T);
    // 4) GEMM2: F = T(1024x1024) · W2(1024x256) + b2
    //    strips = (1024/64)*(256/16) = 256 -> 64 blocks
    wmma_gemm_f32_kernel<false, true, false, SPAD, NDIM, HID, HID, NDIM, NDIM>
        <<<64, 128, 0, stream>>>(T, W2, b2, F);
    // 5) q, C from F
    q_reduce_kernel<<<1, NDIM, 0, stream>>>(F, cs, q, Cr);
    // 6) p, r; overwrite T with E = c_s*(1-T^2)   (after GEMM2 consumed T)
    pr_transform_kernel<<<4, 256, 0, stream>>>(T, cs, p, r);
    // 7) GEMM3: G = Zint^T(256x1024) · E(1024x1024)
    //    strips = (256/64)*(1024/16) = 256 -> 64 blocks
    wmma_gemm_f32_kernel<true, false, false, NDIM, HID, SPAD, NDIM, HID, HID>
        <<<64, 128, 0, stream>>>(Zint, T, nullptr, G);
    // 8) small vectors
    v_kernel<<<1, NDIM, 0, stream>>>(A, g, v);
    w_kernel<<<4, 256, 0, stream>>>(W2, v, w);
    ui_kernel<<<4, 256, 0, stream>>>(z_i, W1, b1, w, ui);
    // 9) assemble outputs
    const int blocks = (out_size + 255) / 256;
    finalize_kernel<<<blocks, 256, 0, stream>>>(A, W1, g, t_i, q, v, w, p, r,
                                                Cr, G, ui, out, out_size);
}